// Encoder_Baseline_TwoStream_28570122453320
// MI455X (gfx1250) — compile-verified
//
#include <hip/hip_runtime.h>
#include <hip/hip_bf16.h>
#include <math.h>

typedef __attribute__((ext_vector_type(2))) float v2f;
typedef __attribute__((ext_vector_type(8))) float v8f;

#define BATCH   64
#define SEQ     512
#define HID     512
#define D0      1536
#define D1      2048
#define M_BIG   (BATCH * SEQ)   // 32768

// ---- CDNA5 async global->LDS (ASYNCcnt-tracked, no VGPR round trip) --------
__device__ __forceinline__ void async_load_b128(const float* gptr, float* ldsptr)
{
    const unsigned lds_addr = (unsigned)(unsigned long long)ldsptr; // low 32b = LDS offset
    asm volatile("global_load_async_to_lds_b128 %0, %1, off"
                 :
                 : "v"(lds_addr), "v"((unsigned long long)gptr)
                 : "memory");
}
#define WAIT_ASYNC(n) asm volatile("s_wait_asynccnt %0" :: "i"(n) : "memory")

// ---------------- tiled GEMM + bias:  C[M,N] = A[M,K] @ W[N,K]^T + b[N] ------
// 128 threads (4 waves), 64x64 tile, K-chunk 64, double-buffered async staging.
#define TM 64
#define TN 64
#define TK 64

__global__ __launch_bounds__(128)
void gemm_bias_f32_wmma(const float* __restrict__ A,
                        const float* __restrict__ W,
                        const float* __restrict__ bias,
                        float* __restrict__ C,
                        int M, int N, int K)
{
    __shared__ float As[2][TM][TK + 4];
    __shared__ float Ws[2][TN][TK + 4];

    const int m0   = blockIdx.x * TM;
    const int n0   = blockIdx.y * TN;
    const int tid  = threadIdx.x;
    const int wave = tid >> 5;          // 0..3
    const int lane = tid & 31;
    const int l15  = lane & 15;
    const int khalf = (lane < 16) ? 0 : 2;

    const int lrow = tid >> 4;          // 0..7 (load pattern)
    const int lcol = (tid & 15) * 4;    // 0..60

    v8f acc[4] = {};                    // 16x64 strip per wave

    // 16 async b128 ops per wave per chunk (8 A + 8 W)
    auto issue_chunk = [&](int buf, int k0) {
#pragma unroll
        for (int i = 0; i < 8; ++i) {
            const int r = lrow + i * 8;
            async_load_b128(A + (size_t)(m0 + r) * K + k0 + lcol, &As[buf][r][lcol]);
            async_load_b128(W + (size_t)(n0 + r) * K + k0 + lcol, &Ws[buf][r][lcol]);
        }
    };

    const int nch = K / TK;
    issue_chunk(0, 0);

    for (int c = 0; c < nch; ++c) {
        const int buf = c & 1;
        if (c + 1 < nch) {
            issue_chunk(buf ^ 1, (c + 1) * TK);  // prefetch next chunk first
            WAIT_ASYNC(16);                      // in-order: chunk c (16 ops) landed
        } else {
            WAIT_ASYNC(0);
        }
        __syncthreads();                         // tile visible to all waves

#pragma unroll
        for (int kk = 0; kk < TK / 4; ++kk) {
            const int kb = kk * 4 + khalf;
            v2f a;
            a.x = As[buf][wave * 16 + l15][kb];
            a.y = As[buf][wave * 16 + l15][kb + 1];
#pragma unroll
            for (int n = 0; n < 4; ++n) {
                v2f b;
                b.x = Ws[buf][n * 16 + l15][kb];
                b.y = Ws[buf][n * 16 + l15][kb + 1];
                acc[n] = __builtin_amdgcn_wmma_f32_16x16x4_f32(
                    false, a, false, b, (short)0, acc[n], false, false);
            }
        }
        __syncthreads();                         // buf reusable for chunk c+2
    }

    // epilogue: bias + store (C layout: VGPR r -> M = r (lanes 0-15) / r+8 (16-31))
#pragma unroll
    for (int n = 0; n < 4; ++n) {
        const int col = n0 + n * 16 + l15;
        const float bv = bias[col];
#pragma unroll
        for (int r = 0; r < 8; ++r) {
            const int row = m0 + wave * 16 + ((lane < 16) ? r : r + 8);
            C[(size_t)row * N + col] = acc[n][r] + bv;
        }
    }
}

// ---------------- init workspace state (graph-replay safe) -------------------
__global__ void gru_init_state(float* hxA, float* hxB, unsigned* cnt)
{
    const int t = blockIdx.x * blockDim.x + threadIdx.x;
    if (t < BATCH * HID) { hxA[t] = 0.0f; hxB[t] = 0.0f; }
    if (t < SEQ) cnt[t] = 0u;
}

// ---------------- fused GRU scan --------------------------------------------
// 32 blocks x 128 threads. Block b owns hidden units [b*16, b*16+16).
// Per step: gh(:, {j, 512+j, 1024+j}) via WMMA (K=512), fused gate math,
// device-wide barrier on a per-step counter, hx ping-pong.
#define NBLK_SCAN 32
#define JT        16

__global__ __launch_bounds__(128)
void gru_scan_wmma(const float* __restrict__ gi,    // [B,S,3H]
                   const float* __restrict__ gf,    // [B,S,2H]
                   const float* __restrict__ Whh,   // [3H,H]
                   const float* __restrict__ bhh,   // [3H]
                   float* __restrict__ hxA,         // [B,H]
                   float* __restrict__ hxB,         // [B,H]
                   unsigned* __restrict__ counters, // [S]
                   float* __restrict__ ys,          // [B,S,H] (d_out)
                   float* __restrict__ hlast)       // [B,H]   (d_out tail)
{
    __shared__ float Wsh[48][HID + 4];   // 3 gates x 16 rows of W_hh, padded

    const int tid  = threadIdx.x;
    const int wave = tid >> 5;           // batch strip: rows wave*16..+15
    const int lane = tid & 31;
    const int l15  = lane & 15;
    const int j0   = blockIdx.x * JT;
    const int khalf = (lane < 16) ? 0 : 2;

    // preload W_hh rows {j0+r, 512+j0+r, 1024+j0+r} into LDS (float4)
    for (int idx = tid; idx < 48 * (HID / 4); idx += 128) {
        const int lr = idx / (HID / 4);            // 0..47
        const int c4 = (idx % (HID / 4)) * 4;      // 0..508
        const int gr = (lr >> 4) * HID + j0 + (lr & 15);
        *(float4*)&Wsh[lr][c4] = *(const float4*)(Whh + (size_t)gr * HID + c4);
    }
    __syncthreads();

    const int   jcol = j0 + l15;
    const float bh_r = bhh[jcol];
    const float bh_i = bhh[jcol + HID];
    const float bh_n = bhh[jcol + 2 * HID];
    const int   mrow = wave * 16 + l15;            // batch row feeding A frags

    for (int s = 0; s < SEQ; ++s) {
        const float* __restrict__ hcur = (s & 1) ? hxB : hxA;
        float* __restrict__       hnxt = (s & 1) ? hxA : hxB;

        v8f hr = {}, hi = {}, hn = {};
        const float* __restrict__ hrow = hcur + (size_t)mrow * HID;

#pragma unroll 4
        for (int kk = 0; kk < HID / 4; ++kk) {
            const int kb = kk * 4 + khalf;
            v2f a;
            a.x = hrow[kb];
            a.y = hrow[kb + 1];
            v2f br, bi, bn;
            br.x = Wsh[l15][kb];        br.y = Wsh[l15][kb + 1];
            bi.x = Wsh[16 + l15][kb];   bi.y = Wsh[16 + l15][kb + 1];
            bn.x = Wsh[32 + l15][kb];   bn.y = Wsh[32 + l15][kb + 1];
            hr = __builtin_amdgcn_wmma_f32_16x16x4_f32(false, a, false, br, (short)0, hr, false, false);
            hi = __builtin_amdgcn_wmma_f32_16x16x4_f32(false, a, false, bi, (short)0, hi, false, false);
            hn = __builtin_amdgcn_wmma_f32_16x16x4_f32(false, a, false, bn, (short)0, hn, false, false);
        }

        // fused gate math; lane/r -> (batch bb, hidden jcol)
#pragma unroll
        for (int r = 0; r < 8; ++r) {
            const int bb = wave * 16 + ((lane < 16) ? r : r + 8);
            const size_t bs   = (size_t)bb * SEQ + s;
            const size_t ogi  = bs * (3 * HID) + jcol;
            const size_t ogf  = bs * (2 * HID) + jcol;
            const float i_r = gi[ogi];
            const float i_i = gi[ogi + HID];
            const float i_n = gi[ogi + 2 * HID];
            const float f_r = gf[ogf];
            const float f_i = gf[ogf + HID];
            const float hprev = hcur[(size_t)bb * HID + jcol];

            const float rg = 1.0f / (1.0f + __expf(-(i_r + hr[r] + bh_r + f_r)));
            const float ig = 1.0f / (1.0f + __expf(-(i_i + hi[r] + bh_i + f_i)));
            const float ng = tanhf(i_n + rg * (hn[r] + bh_n));
            const float hy = ng + ig * (hprev - ng);

            hnxt[(size_t)bb * HID + jcol] = hy;
            ys[bs * HID + jcol] = hy;
            if (s == SEQ - 1) hlast[(size_t)bb * HID + jcol] = hy;
        }

        // device-wide barrier (fresh counter per step)
        __threadfence();
        __syncthreads();
        if (tid == 0) {
            atomicAdd(&counters[s], 1u);
            while (__hip_atomic_load(&counters[s], __ATOMIC_ACQUIRE,
                                     __HIP_MEMORY_SCOPE_AGENT) < (unsigned)NBLK_SCAN) {
                __builtin_amdgcn_s_sleep(1);
            }
        }
        __syncthreads();
        __threadfence();
    }
}

// ---------------- launch -----------------------------------------------------
extern "C" void kernel_launch(void* const* d_in, const int* in_sizes, int n_in,
                              void* d_out, int out_size, void* d_ws, size_t ws_size,
                              hipStream_t stream)
{
    const float* feats0 = (const float*)d_in[0];   // [64,512,1536]
    const float* feats1 = (const float*)d_in[1];   // [64,512,2048]
    const float* W0     = (const float*)d_in[2];   // [1024,1536]
    const float* b0     = (const float*)d_in[3];
    const float* W1     = (const float*)d_in[4];   // [1536,2048]
    const float* b1     = (const float*)d_in[5];
    const float* Whh    = (const float*)d_in[6];   // [1536,512]
    const float* bhh    = (const float*)d_in[7];

    float* gi  = (float*)d_ws;                         // [64,512,1536]
    float* gf  = gi + (size_t)M_BIG * 3 * HID;         // [64,512,1024]
    float* hxA = gf + (size_t)M_BIG * 2 * HID;         // [64,512]
    float* hxB = hxA + BATCH * HID;                    // [64,512]
    unsigned* cnt = (unsigned*)(hxB + BATCH * HID);    // [512]

    float* ys    = (float*)d_out;                      // [64,512,512]
    float* hlast = ys + (size_t)BATCH * SEQ * HID;     // [64,512]

    // reset scan state each launch (graph-replay deterministic)
    gru_init_state<<<(BATCH * HID + 255) / 256, 256, 0, stream>>>(hxA, hxB, cnt);

    // gi = feats1 @ W1^T + b1   (M=32768, N=1536, K=2048)
    gemm_bias_f32_wmma<<<dim3(M_BIG / TM, (3 * HID) / TN), 128, 0, stream>>>(
        feats1, W1, b1, gi, M_BIG, 3 * HID, D1);

    // gf = feats0 @ W0^T + b0   (M=32768, N=1024, K=1536)
    gemm_bias_f32_wmma<<<dim3(M_BIG / TM, (2 * HID) / TN), 128, 0, stream>>>(
        feats0, W0, b0, gf, M_BIG, 2 * HID, D0);

    // fused sequential GRU scan
    gru_scan_wmma<<<NBLK_SCAN, 128, 0, stream>>>(
        gi, gf, Whh, bhh, hxA, hxB, cnt, ys, hlast);
}